// DRCapsuleCONV_79817672229003
// MI455X (gfx1250) — compile-verified
//
#include <hip/hip_runtime.h>
#include <math.h>

typedef float v2f __attribute__((ext_vector_type(2)));
typedef float v8f __attribute__((ext_vector_type(8)));

#define NIN    32
#define MOUT   32
#define DOUT   16
#define WOUT   15
#define HWOUT  225      // 15*15
#define CDIM   4608     // NIN * 9 * 16  (contraction length)
#define COLS   512      // MOUT * DOUT
#define ROWS   16       // spatial positions per workgroup (WMMA M)
#define NBATCH 225      // 3600 positions / 16
#define NSTEPS 1152     // CDIM / 4  (WMMA K-steps)

// x (B=16, N=32, H=32, W=32, A=16) strides in floats
#define XS_B 524288
#define XS_N 16384
#define XS_H 512
#define XS_W 16

// nqk (B, N, 3, 3, M, 15, 15) strides in floats
#define QS_B 2073600
#define QS_N 64800
#define QS_KL 7200
#define QS_M 225

#define NQK_ELEMS 33177600
#define WPK_ELEMS 2359296   // NSTEPS * 32 tiles * 64 floats (== |w|, a permutation)

// XOR swizzle so 16 rows (stride 4608 = 0 mod 64 banks) don't collide on
// A-fragment ds_load_b64; toggles bits 1..3 only -> dword pair stays
// contiguous and 8B-aligned.
__device__ __forceinline__ int idxU(int row, int c) {
    return row * CDIM + (c ^ ((row & 7) << 1));
}

#define WMMA4(ACC, A, B) \
    ACC = __builtin_amdgcn_wmma_f32_16x16x4_f32(false, A, false, B, (short)0, ACC, false, false)

// ---------------------------------------------------------------------------
// Repack w into per-WMMA B-fragment order:
//   wpk[((S*32 + m)*32 + lane)*2 + j] = B-frag element (VGPR j) of lane for
//   K-step S and m-tile m.  B layout: VGPR j holds row 2*(lane>>4)+j, col lane&15.
// ---------------------------------------------------------------------------
__global__ __launch_bounds__(256)
void pack_w_kernel(const float* __restrict__ w, float* __restrict__ wpk)
{
    const int idx = blockIdx.x * 256 + threadIdx.x;   // one (S, m, lane) per thread
    const int l   = idx & 31;
    const int m   = (idx >> 5) & 31;
    const int S   = idx >> 10;                        // 0..1151
    const int s   = S & 3;
    const int nkl = S >> 2;
    const int n   = nkl / 9;
    const int kl  = nkl - n * 9;
    const int rowb = (kl * 32 + n) * 16 + s * 4 + ((l >> 4) << 1);
    const int col  = m * 16 + (l & 15);
    v2f frag;
    frag.x = w[rowb * COLS + col];
    frag.y = w[(rowb + 1) * COLS + col];
    *(v2f*)&wpk[idx * 2] = frag;
}

// ---------------------------------------------------------------------------
// Fused: unfold -> (phase A GEMM, squash, LayerNorm, out) -> (phase B GEMM,
// d-contraction with v, nqk)
// ---------------------------------------------------------------------------
template <bool PACKED>
__global__ __launch_bounds__(256)
void drcaps_fused_kernel(const float* __restrict__ x,
                         const float* __restrict__ wsrc,   // packed (PACKED) or raw w
                         const float* __restrict__ gamma,
                         const float* __restrict__ beta,
                         float* __restrict__ nqk,
                         float* __restrict__ outln)
{
    extern __shared__ float smem[];
    float* uS = smem;                  // 16 x 4608 floats (288 KB)
    float* vS = smem + ROWS * CDIM;    // 16 x 512 floats (32 KB); tail reused as stage

    const int tid  = threadIdx.x;
    const int lane = tid & 31;
    const int wave = tid >> 5;          // 0..7
    const int half = lane >> 4;         // 0 or 1
    const int ln16 = lane & 15;
    const int wg   = blockIdx.x;        // 0..224

    // ---------------- gather unfolded input rows into LDS ----------------
    #pragma unroll 1
    for (int r = 0; r < ROWS; ++r) {
        const int P  = wg * ROWS + r;
        const int b  = P / HWOUT;
        const int hw = P - b * HWOUT;
        const int h  = hw / WOUT;
        const int ww = hw - h * WOUT;
        const int base = b * XS_B + (2 * h) * XS_H + (2 * ww) * XS_W;
        #pragma unroll 1
        for (int c = tid; c < CDIM; c += 256) {
            const int a  = c & 15;
            const int t2 = c >> 4;             // n*9 + kl
            const int n  = t2 / 9;
            const int kl = t2 - n * 9;
            const int k  = kl / 3;
            const int l  = kl - k * 3;
            uS[idxU(r, c)] = x[base + n * XS_N + k * XS_H + l * XS_W + a];
        }
    }
    __syncthreads();

    const int koff = half * 2;          // A frag: VGPR0/1 hold K = koff, koff+1
    const int mb   = wave * 4;          // this wave's 4 m-tiles

    // B-fragment loader for the 4 m-tiles at K-step S = (n*9+kl)*4 + s
    auto loadB4 = [&](int n, int kl, int s, v2f& B0, v2f& B1, v2f& B2, v2f& B3) {
        if constexpr (PACKED) {
            const int S = (n * 9 + kl) * 4 + s;
            const float* wf = wsrc + S * 2048 + lane * 2;   // 2048 = 32 tiles * 64
            B0 = *(const v2f*)&wf[(mb + 0) * 64];
            B1 = *(const v2f*)&wf[(mb + 1) * 64];
            B2 = *(const v2f*)&wf[(mb + 2) * 64];
            B3 = *(const v2f*)&wf[(mb + 3) * 64];
        } else {
            const int wr = ((kl * 32 + n) * 16 + s * 4 + koff) * COLS + ln16;
            const float* w0 = wsrc + wr + (mb + 0) * 16;
            const float* w1 = wsrc + wr + (mb + 1) * 16;
            const float* w2 = wsrc + wr + (mb + 2) * 16;
            const float* w3 = wsrc + wr + (mb + 3) * 16;
            B0.x = w0[0]; B0.y = w0[COLS];
            B1.x = w1[0]; B1.y = w1[COLS];
            B2.x = w2[0]; B2.y = w2[COLS];
            B3.x = w3[0]; B3.y = w3[COLS];
        }
    };

    // ---------------- phase A: V = (1/32) * U(16x4608) x W(4608x512) -----
    v8f acc0 = {}, acc1 = {}, acc2 = {}, acc3 = {};
    #pragma unroll 1
    for (int n = 0; n < NIN; ++n) {
        #pragma unroll 1
        for (int kl = 0; kl < 9; ++kl) {
            if constexpr (PACKED) {
                const int Snext = (n * 9 + kl) * 4 + 4;     // next kl block
                if (Snext < NSTEPS)
                    __builtin_prefetch(wsrc + Snext * 2048 + tid * 32, 0, 1);
            }
            #pragma unroll
            for (int s = 0; s < 4; ++s) {
                const int c0 = (n * 9 + kl) * 16 + s * 4;
                v2f A = *(const v2f*)&uS[idxU(ln16, c0 + koff)];
                v2f B0, B1, B2, B3;
                loadB4(n, kl, s, B0, B1, B2, B3);
                WMMA4(acc0, A, B0);
                WMMA4(acc1, A, B1);
                WMMA4(acc2, A, B2);
                WMMA4(acc3, A, B3);
            }
        }
    }
    // D layout: lanes0-15 = cols, VGPR r = row r (lanes0-15) / r+8 (lanes16-31)
    #pragma unroll
    for (int r = 0; r < 8; ++r) {
        const int pos = r + half * 8;
        vS[pos * COLS + (mb + 0) * 16 + ln16] = acc0[r] * 0.03125f;
        vS[pos * COLS + (mb + 1) * 16 + ln16] = acc1[r] * 0.03125f;
        vS[pos * COLS + (mb + 2) * 16 + ln16] = acc2[r] * 0.03125f;
        vS[pos * COLS + (mb + 3) * 16 + ln16] = acc3[r] * 0.03125f;
    }
    __syncthreads();

    // ---------------- squash + LayerNorm, write `out` ---------------------
    #pragma unroll
    for (int qi = 0; qi < 2; ++qi) {
        const int q   = tid + qi * 256;     // pos*32 + m
        const int pos = q >> 5;
        const int m   = q & 31;
        float* vp = &vS[pos * COLS + m * 16];
        float vv[16];
        float sq = 0.f;
        #pragma unroll
        for (int d = 0; d < 16; ++d) { vv[d] = vp[d]; sq += vv[d] * vv[d]; }
        const float scale = sqrtf(sq) / (1.0f + sq);   // sq/(1+sq) / sqrt(sq)
        float mu = 0.f;
        #pragma unroll
        for (int d = 0; d < 16; ++d) { vv[d] *= scale; vp[d] = vv[d]; mu += vv[d]; }
        mu *= 0.0625f;
        float var = 0.f;
        #pragma unroll
        for (int d = 0; d < 16; ++d) { const float t = vv[d] - mu; var += t * t; }
        var *= 0.0625f;
        const float inv = rsqrtf(var + 1e-5f);
        const int P  = wg * ROWS + pos;
        const int b  = P / HWOUT;
        const int hw = P - b * HWOUT;
        float* op = outln + ((b * MOUT + m) * HWOUT + hw) * DOUT;
        #pragma unroll
        for (int d = 0; d < 16; ++d) op[d] = (vv[d] - mu) * inv * gamma[d] + beta[d];
    }
    __syncthreads();

    // ---------------- load squashed-v fragments (C-layout) ----------------
    v8f vf0, vf1, vf2, vf3;
    #pragma unroll
    for (int r = 0; r < 8; ++r) {
        const int pos = r + half * 8;
        vf0[r] = vS[pos * COLS + (mb + 0) * 16 + ln16];
        vf1[r] = vS[pos * COLS + (mb + 1) * 16 + ln16];
        vf2[r] = vS[pos * COLS + (mb + 2) * 16 + ln16];
        vf3[r] = vS[pos * COLS + (mb + 3) * 16 + ln16];
    }
    __syncthreads();   // vS free for reuse as double-buffered nqk stage

    // ---------------- phase B: Z = U_nkl x W_nkl, nqk = sum_d Z*v ---------
    const int posT = tid & 15;
    const int mT   = tid >> 4;          // 0..15
    const int PT   = wg * ROWS + posT;
    const int bT   = PT / HWOUT;
    const int hwT  = PT - bT * HWOUT;
    const int baseT0 = bT * QS_B + (mT)      * QS_M + hwT;
    const int baseT1 = bT * QS_B + (mT + 16) * QS_M + hwT;

    #pragma unroll 1
    for (int n = 0; n < NIN; ++n) {
        #pragma unroll 1
        for (int kl = 0; kl < 9; ++kl) {
            if constexpr (PACKED) {
                const int Snext = (n * 9 + kl) * 4 + 4;
                if (Snext < NSTEPS)
                    __builtin_prefetch(wsrc + Snext * 2048 + tid * 32, 0, 1);
            }
            v8f z0 = {}, z1 = {}, z2 = {}, z3 = {};
            #pragma unroll
            for (int s = 0; s < 4; ++s) {
                const int c0 = (n * 9 + kl) * 16 + s * 4;
                v2f A = *(const v2f*)&uS[idxU(ln16, c0 + koff)];
                v2f B0, B1, B2, B3;
                loadB4(n, kl, s, B0, B1, B2, B3);
                WMMA4(z0, A, B0);
                WMMA4(z1, A, B1);
                WMMA4(z2, A, B2);
                WMMA4(z3, A, B3);
            }
            // double-buffered stage (one barrier per iteration)
            float* stage = vS + ((n * 9 + kl) & 1) * COLS;
            // fold over d (lanes 0..15 of each half) with xor-shuffle tree
            #pragma unroll
            for (int r = 0; r < 8; ++r) {
                float p0 = z0[r] * vf0[r];
                float p1 = z1[r] * vf1[r];
                float p2 = z2[r] * vf2[r];
                float p3 = z3[r] * vf3[r];
                #pragma unroll
                for (int msk = 1; msk <= 8; msk <<= 1) {
                    p0 += __shfl_xor(p0, msk, 32);
                    p1 += __shfl_xor(p1, msk, 32);
                    p2 += __shfl_xor(p2, msk, 32);
                    p3 += __shfl_xor(p3, msk, 32);
                }
                if (ln16 == 0) {
                    float* st = &stage[(r + half * 8) * 32];  // stage[pos][m]
                    st[mb + 0] = p0;
                    st[mb + 1] = p1;
                    st[mb + 2] = p2;
                    st[mb + 3] = p3;
                }
            }
            __syncthreads();
            // coalesced store: consecutive pos -> consecutive hw (inner dim)
            const int off = n * QS_N + kl * QS_KL;
            nqk[baseT0 + off] = stage[posT * 32 + mT];
            nqk[baseT1 + off] = stage[posT * 32 + mT + 16];
            // no second barrier: next iteration writes the other stage buffer
        }
    }
}

extern "C" void kernel_launch(void* const* d_in, const int* in_sizes, int n_in,
                              void* d_out, int out_size, void* d_ws, size_t ws_size,
                              hipStream_t stream) {
    (void)in_sizes; (void)n_in; (void)out_size;
    const float* x     = (const float*)d_in[0];
    const float* w     = (const float*)d_in[1];
    const float* gamma = (const float*)d_in[2];
    const float* beta  = (const float*)d_in[3];
    float* nqk   = (float*)d_out;
    float* outln = nqk + NQK_ELEMS;

    const size_t lds_bytes = (size_t)(ROWS * CDIM + ROWS * COLS) * sizeof(float); // 320 KB
    const size_t pk_bytes  = (size_t)WPK_ELEMS * sizeof(float);                   // ~9.4 MB

    if (d_ws != nullptr && ws_size >= pk_bytes) {
        float* wpk = (float*)d_ws;
        pack_w_kernel<<<dim3(WPK_ELEMS / 2 / 256), dim3(256), 0, stream>>>(w, wpk);
        (void)hipFuncSetAttribute(reinterpret_cast<const void*>(drcaps_fused_kernel<true>),
                                  hipFuncAttributeMaxDynamicSharedMemorySize, (int)lds_bytes);
        drcaps_fused_kernel<true><<<dim3(NBATCH), dim3(256), lds_bytes, stream>>>(
            x, wpk, gamma, beta, nqk, outln);
    } else {
        (void)hipFuncSetAttribute(reinterpret_cast<const void*>(drcaps_fused_kernel<false>),
                                  hipFuncAttributeMaxDynamicSharedMemorySize, (int)lds_bytes);
        drcaps_fused_kernel<false><<<dim3(NBATCH), dim3(256), lds_bytes, stream>>>(
            x, w, gamma, beta, nqk, outln);
    }
}